// MagnitudeGradientLoss_82695300317218
// MI455X (gfx1250) — compile-verified
//
#include <hip/hip_runtime.h>
#include <math.h>
#include <stdint.h>

// Problem geometry (fixed by the reference).
#define HH 768
#define WW 768
#define NIMG 64
#define INNER 766            // interior rows/cols
#define TILE_H 16            // output rows per block
#define TILES_PER_IMG 48     // ceil(766 / 16)
#define NBLOCKS (NIMG * TILES_PER_IMG)
#define INV2W (0.5f / 0.0607f)
#define EPSF 1e-8f

// ---------------------------------------------------------------------------
// Kernel 1: stage an 18-row x 768-col tile into LDS via CDNA5 async
// global->LDS copies (ASYNCcnt path), compute the 5-point stencil from LDS,
// reduce per-block, write one partial per block.
// ---------------------------------------------------------------------------
__global__ __launch_bounds__(256) void stencil_partial(const float* __restrict__ phi,
                                                       float* __restrict__ partial) {
    // 18 rows * 768 cols * 4B = 55,296 B of LDS, 16B aligned for b128 async loads.
    __shared__ __align__(16) float s_tile[(TILE_H + 2) * WW];
    __shared__ float wsum[8];

    const int tid  = threadIdx.x;
    const int bid  = blockIdx.x;
    const int img  = bid / TILES_PER_IMG;
    const int tile = bid - img * TILES_PER_IMG;

    const int r0        = tile * TILE_H;                 // first staged global row
    const int rowsStage = min(TILE_H + 2, HH - r0);      // rows copied into LDS
    const int rowsOut   = min(TILE_H, INNER - r0);       // interior rows produced

    const float* gbase = phi + (size_t)img * HH * WW + (size_t)r0 * WW;

    // Low 32 bits of the flat shared-memory pointer == LDS byte offset
    // (LDS aperture occupies addr[63:32] only).
    unsigned lds_base = (unsigned)(uintptr_t)(void*)s_tile;

    // Async-stage the tile: 192 float4 per row, per-lane b128 copies.
    const int totalVec = rowsStage * (WW / 4);
    for (int i = tid; i < totalVec; i += 256) {
        unsigned     lds = lds_base + (unsigned)(i * 16);
        const float* g   = gbase + (size_t)i * 4;
        asm volatile("global_load_async_to_lds_b128 %0, %1, off"
                     :: "v"(lds), "v"(g)
                     : "memory");
    }
    // Wait for this wave's async copies, then barrier so every wave sees the tile.
    asm volatile("s_wait_asynccnt 0x0" ::: "memory");
    __syncthreads();

    // Stencil from LDS: center row lr+1, neighbors lr and lr+2.
    float acc = 0.0f;
    for (int lr = 0; lr < rowsOut; ++lr) {
        const float* rm = &s_tile[lr * WW];   // row above
        const float* rc = rm + WW;            // center row
        const float* rp = rc + WW;            // row below
        for (int c = 1 + tid; c <= INNER; c += 256) {
            float dx = (rc[c + 1] - rc[c - 1]) * INV2W;
            float dy = (rp[c] - rm[c]) * INV2W;
            acc += sqrtf(dx * dx + dy * dy + EPSF);
        }
    }

    // wave32 shuffle reduction, then cross-wave via LDS.
    for (int off = 16; off > 0; off >>= 1)
        acc += __shfl_down(acc, off, 32);
    const int lane = tid & 31;
    const int wid  = tid >> 5;
    if (lane == 0) wsum[wid] = acc;
    __syncthreads();
    if (tid == 0) {
        float s = 0.0f;
        for (int i = 0; i < 8; ++i) s += wsum[i];
        partial[bid] = s;
    }
}

// ---------------------------------------------------------------------------
// Kernel 2: deterministic final reduction of the 3072 block partials and the
// mean division (avoids non-deterministic float atomics).
// ---------------------------------------------------------------------------
__global__ __launch_bounds__(256) void final_reduce(const float* __restrict__ partial,
                                                    float* __restrict__ out) {
    __shared__ float wsum[8];
    float acc = 0.0f;
    for (int i = threadIdx.x; i < NBLOCKS; i += 256)
        acc += partial[i];
    for (int off = 16; off > 0; off >>= 1)
        acc += __shfl_down(acc, off, 32);
    const int lane = threadIdx.x & 31;
    const int wid  = threadIdx.x >> 5;
    if (lane == 0) wsum[wid] = acc;
    __syncthreads();
    if (threadIdx.x == 0) {
        float s = 0.0f;
        for (int i = 0; i < 8; ++i) s += wsum[i];
        const float n = (float)((long long)NIMG * INNER * INNER);  // 37,552,384 (exact in f32)
        out[0] = s / (n + EPSF);
    }
}

extern "C" void kernel_launch(void* const* d_in, const int* in_sizes, int n_in,
                              void* d_out, int out_size, void* d_ws, size_t ws_size,
                              hipStream_t stream) {
    const float* phi     = (const float*)d_in[0];
    float*       out     = (float*)d_out;
    float*       partial = (float*)d_ws;   // NBLOCKS floats = 12 KB of scratch

    stencil_partial<<<NBLOCKS, 256, 0, stream>>>(phi, partial);
    final_reduce<<<1, 256, 0, stream>>>(partial, out);
}